// L2HistogramSingularityStrengthRecalibration_85409719648437
// MI455X (gfx1250) — compile-verified
//
#include <hip/hip_runtime.h>
#include <stdint.h>

// L2HistogramSingularityStrengthRecalibration for MI455X (gfx1250).
//
// Roofline: 12.85M f32 elements; min HBM traffic = 51.4MB read + 51.4MB write
// = ~103MB -> ~4.4us at 23.3 TB/s. Arithmetic ~60 f32 VALU ops + 5
// transcendentals per element -> bandwidth/VALU bound. No matmul exists in
// this op (the ReLU in the histogram kills any factorization), so WMMA is
// deliberately absent: a banded-matmul box filter on V_WMMA_F32_16X16X4_F32
// wastes >6x flops on structural zeros on the slow f32 matrix path. The
// CDNA5-specific feature that fits is the async data mover:
// GLOBAL_LOAD_ASYNC_TO_LDS_B128 staging + s_wait_asynccnt, with NT stores
// for the write-once output so L2 keeps the x halo working set.

#define EPSV 1e-6f

constexpr int Bn   = 16;
constexpr int Hn   = 112;
constexpr int Wn   = 112;
constexpr int Cn   = 64;
constexpr int Kn   = 8;
constexpr int TILE = 14;            // output tile side (112 = 8*14)
constexpr int RMAX = 4;             // max pooling radius
constexpr int TS   = TILE + 2*RMAX; // 22 haloed tile side
constexpr int CCH  = 32;            // channels per chunk (== wave32 lanes)

__global__ __launch_bounds__(256)
void lhssr_kernel(const float* __restrict__ x,
                  const float* __restrict__ sw,
                  const float* __restrict__ centers,
                  const float* __restrict__ widths,
                  float* __restrict__ out)
{
    // Haloed tile, layout T[(h*TS + w)*CCH + c]; 22*22*32*4 = 61,952 B.
    // 320KB/WGP => up to 5 workgroups (40 waves) resident per WGP.
    __shared__ __align__(16) float T[TS * TS * CCH];

    const int t    = (int)threadIdx.x;
    const int lane = t & 31;
    const int wave = t >> 5;

    const int b     = (int)blockIdx.y;
    const int chunk = (int)blockIdx.z;
    const int h0    = ((int)blockIdx.x >> 3) * TILE;  // 8 tiles per row
    const int w0    = ((int)blockIdx.x & 7)  * TILE;
    const int c0    = chunk * CCH;

    // ---------------- Phase 0: async DMA raw x tile -> LDS -----------------
    // Per-lane 16B transfers, GV addressing (64-bit vaddr), tracked on
    // ASYNCcnt. Out-of-image positions are simply not fetched; Phase A gates
    // them out, so no zero-fill pass is needed.
    {
        const int NF4 = TS * TS * (CCH / 4);  // 3872 float4 transfers
        for (int idx = t; idx < NF4; idx += 256) {
            const int c4   = idx & 7;
            const int wcol = (idx >> 3) % TS;
            const int hrow = (idx >> 3) / TS;
            const int gh   = h0 - RMAX + hrow;
            const int gw   = w0 - RMAX + wcol;
            if ((unsigned)gh < (unsigned)Hn && (unsigned)gw < (unsigned)Wn) {
                const float* gp = x + (((size_t)b * Hn + gh) * Wn + gw) * Cn
                                    + c0 + c4 * 4;
                // Low 32 bits of a flat LDS pointer == wave-relative LDS offset.
                const uint32_t lds = (uint32_t)(uintptr_t)(&T[idx * 4]);
                asm volatile("global_load_async_to_lds_b128 %0, %1, off"
                             :: "v"(lds), "v"(gp) : "memory");
            }
        }
        asm volatile("s_wait_asynccnt 0" ::: "memory");
    }
    __syncthreads();

    // ------------- Phase A: horizontal prefix of xa=|x|+eps (in place) -----
    // EPS gated so out-of-image padding contributes exactly 0 (matches
    // lax.reduce_window SAME zero padding). Lanes = channels -> every LDS
    // access hits 32 consecutive banks (conflict-free).
    for (int s = t; s < TS * CCH; s += 256) {
        const int h  = s >> 5;          // CCH == 32
        const int c  = s & 31;
        const int gh = h0 - RMAX + h;
        const bool rowOK = (unsigned)gh < (unsigned)Hn;
        const int base = h * TS * CCH + c;
        float acc = 0.0f;
        for (int w = 0; w < TS; ++w) {
            const int gw = w0 - RMAX + w;
            if (rowOK && (unsigned)gw < (unsigned)Wn)
                acc += fabsf(T[base + w * CCH]) + EPSV;
            T[base + w * CCH] = acc;
        }
    }
    __syncthreads();

    // ------------- Phase B: vertical prefix (in place) ---------------------
    for (int s = t; s < TS * CCH; s += 256) {
        const int w = s >> 5;
        const int c = s & 31;
        const int base = w * CCH + c;
        float acc = 0.0f;
        for (int h = 0; h < TS; ++h) {
            acc += T[base + h * TS * CCH];
            T[base + h * TS * CCH] = acc;
        }
    }
    __syncthreads();

    // ------------- Phase C: box sums via 4 corners, log/dot/hist/sigmoid ---
    const int c = c0 + lane;
    const float4 ctr0 = ((const float4*)(centers + (size_t)c * Kn))[0];
    const float4 ctr1 = ((const float4*)(centers + (size_t)c * Kn))[1];
    const float4 wd0  = ((const float4*)(widths  + (size_t)c * Kn))[0];
    const float4 wd1  = ((const float4*)(widths  + (size_t)c * Kn))[1];
    const float s0 = sw[0], s1 = sw[1], s2 = sw[2], s3 = sw[3];

    // Pixel base for this block: &x/out[b][h0][w0][c]
    const size_t tileBase = (((size_t)b * Hn + h0) * Wn + w0) * Cn + c;

    // Incremental pr/pc (avoid per-iteration div/mod: wave-strided by 8).
    int pr = 0, pc = wave;
    for (int p = wave; p < TILE * TILE; p += 8) {
        const int tr = RMAX + pr;
        const int tc = RMAX + pc;

        float lm[RMAX];
        #pragma unroll
        for (int r = 1; r <= RMAX; ++r) {
            const int i1 = tr + r, j1 = tc + r;          // <= TS-1 always
            const int i0 = tr - r - 1, j0 = tc - r - 1;  // >= -1
            const float a  =                T[(i1 * TS + j1) * CCH + lane];
            const float bb = (i0 >= 0) ?    T[(i0 * TS + j1) * CCH + lane] : 0.0f;
            const float cc = (j0 >= 0) ?    T[(i1 * TS + j0) * CCH + lane] : 0.0f;
            const float dd = (i0 >= 0 && j0 >= 0)
                                       ?    T[(i0 * TS + j0) * CCH + lane] : 0.0f;
            const float mu = a - bb - cc + dd;           // > 0 (interior pixel)
            lm[r - 1] = __logf(mu);
        }
        const float alpha = s0 * lm[0] + s1 * lm[1] + s2 * lm[2] + s3 * lm[3];

        float scnt = 0.0f, d;
        d = alpha - ctr0.x; scnt += fmaxf(fmaf(-wd0.x, d * d, 1.0f), 0.0f);
        d = alpha - ctr0.y; scnt += fmaxf(fmaf(-wd0.y, d * d, 1.0f), 0.0f);
        d = alpha - ctr0.z; scnt += fmaxf(fmaf(-wd0.z, d * d, 1.0f), 0.0f);
        d = alpha - ctr0.w; scnt += fmaxf(fmaf(-wd0.w, d * d, 1.0f), 0.0f);
        d = alpha - ctr1.x; scnt += fmaxf(fmaf(-wd1.x, d * d, 1.0f), 0.0f);
        d = alpha - ctr1.y; scnt += fmaxf(fmaf(-wd1.y, d * d, 1.0f), 0.0f);
        d = alpha - ctr1.z; scnt += fmaxf(fmaf(-wd1.z, d * d, 1.0f), 0.0f);
        d = alpha - ctr1.w; scnt += fmaxf(fmaf(-wd1.w, d * d, 1.0f), 0.0f);

        const float sig = 1.0f / (1.0f + __expf(-scnt));

        // x reload hits L2 (async fill already pulled it through L2; entire
        // x is 51MB of 192MB L2). Output is write-once: stream it NT so it
        // does not evict the halo working set.
        const size_t gidx = tileBase + ((size_t)(pr * Wn + pc)) * Cn;
        __builtin_nontemporal_store(x[gidx] + sig, &out[gidx]);

        pc += 8;
        if (pc >= TILE) { pc -= TILE; ++pr; }
    }
}

extern "C" void kernel_launch(void* const* d_in, const int* in_sizes, int n_in,
                              void* d_out, int out_size, void* d_ws, size_t ws_size,
                              hipStream_t stream) {
    const float* x       = (const float*)d_in[0];
    const float* sw      = (const float*)d_in[1];
    const float* centers = (const float*)d_in[2];
    const float* widths  = (const float*)d_in[3];
    float* out           = (float*)d_out;

    dim3 grid((Hn / TILE) * (Wn / TILE), Bn, Cn / CCH);  // 64 x 16 x 2
    lhssr_kernel<<<grid, 256, 0, stream>>>(x, sw, centers, widths, out);
}